// AnchorHeadPrune_59124519797212
// MI455X (gfx1250) — compile-verified
//
#include <hip/hip_runtime.h>

typedef float v2f __attribute__((ext_vector_type(2)));
typedef float v4f __attribute__((ext_vector_type(4)));
typedef float v8f __attribute__((ext_vector_type(8)));

#define NVOX   400000
#define KDIM   128
#define NCOL   108            // 60 cls + 42 reg + 6 obj (output column order)
#define PITCH  132            // LDS row pitch in floats: 132 mod 64 = 4 -> conflict-free ds_load_b64
#define TPB    8              // 16-row tiles per block
#define NTILES (NVOX / 16)    // 25000
#define NBLK   (NTILES / TPB) // 3125 (exact)

// ---------------- CDNA5 async global->LDS path (guarded) ----------------
#if __has_builtin(__builtin_amdgcn_global_load_async_to_lds_b128) && \
    __has_builtin(__builtin_amdgcn_s_wait_asynccnt)
#define USE_ASYNC_LDS 1
// Builtin signature (from compiler diagnostic): param type is
// 'int __attribute__((vector_size(16)))  *' in the global / LDS address spaces.
typedef int v4i_gcc __attribute__((vector_size(16)));
typedef __attribute__((address_space(1))) v4i_gcc* g_b128_p;
typedef __attribute__((address_space(3))) v4i_gcc* l_b128_p;
__device__ __forceinline__ void async_cp16(const float* g, float* l) {
    // generic LDS addr low 32 bits == LDS byte offset (aperture truncation rule)
    __builtin_amdgcn_global_load_async_to_lds_b128(
        (g_b128_p)(uintptr_t)g, (l_b128_p)(uint32_t)(uintptr_t)l, 0, 0);
}
#else
#define USE_ASYNC_LDS 0
#endif

// Exact-fp32 matrix op: D = A(16x4) * B(4x16) + C(16x16)
__device__ __forceinline__ v8f wmma_f32x4(v2f a, v2f b, v8f c) {
    return __builtin_amdgcn_wmma_f32_16x16x4_f32(false, a, false, b, (short)0, c,
                                                 false, false);
}

struct ColMap {
    const float* wb;   // &W[0][col] in the owning weight array
    float*       ob;   // &out[0*stride + col] in the owning output region
    float        bias;
    int          ws;   // weight row stride
    int          os;   // output row stride
    bool         val;  // column < 108
};

__device__ __forceinline__ ColMap map_col(int n,
                                          const float* Wc, const float* bc,
                                          const float* Wo, const float* bo,
                                          const float* Wr, const float* br,
                                          float* out) {
    ColMap cm;
    if (n < 60) {                        // cls region: out[row*60 + n]
        cm.wb = Wc + n;  cm.ws = 60; cm.bias = bc[n];
        cm.ob = out + n; cm.os = 60; cm.val = true;
    } else if (n < 102) {                // box_reg region at NVOX*60
        int c = n - 60;
        cm.wb = Wr + c;  cm.ws = 42; cm.bias = br[c];
        cm.ob = out + (size_t)NVOX * 60 + c; cm.os = 42; cm.val = true;
    } else if (n < 108) {                // objness region at NVOX*102
        int c = n - 102;
        cm.wb = Wo + c;  cm.ws = 6;  cm.bias = bo[c];
        cm.ob = out + (size_t)NVOX * 102 + c; cm.os = 6; cm.val = true;
    } else {                             // padding column (never stored)
        cm.wb = Wc; cm.ws = 0; cm.bias = 0.f; cm.ob = out; cm.os = 0; cm.val = false;
    }
    return cm;
}

__global__ __launch_bounds__(128, 1)
void anchor_head_fused(const float* __restrict__ x,
                       const float* __restrict__ Wc, const float* __restrict__ bc,
                       const float* __restrict__ Wo, const float* __restrict__ bo,
                       const float* __restrict__ Wr, const float* __restrict__ br,
                       float* __restrict__ out)
{
    __shared__ float lds[2][16 * PITCH];   // double-buffered 16x128 x-tile, padded

    const int tid  = threadIdx.x;
    const int lane = tid & 31;             // wave32
    const int wave = tid >> 5;             // 4 waves/block
    const int h    = lane >> 4;            // lane half
    const int m    = lane & 15;            // A: row, B/C/D: column within tile

    // ---- per-lane column mapping for this wave's two 16-col tiles ----
    const ColMap c0 = map_col((wave * 2 + 0) * 16 + m, Wc, bc, Wo, bo, Wr, br, out);
    const ColMap c1 = map_col((wave * 2 + 1) * 16 + m, Wc, bc, Wo, bo, Wr, br, out);

    // ---- preload B fragments into registers (amortized over TPB row tiles) ----
    // B 4x16 f32: VGPR0 = row k (lanes0-15) / k+2 (lanes16-31); VGPR1 = k+1 / k+3.
    // Dead columns get B == 0, so the corresponding WMMA lanes are harmless.
    v2f B0[32], B1[32];
#pragma unroll
    for (int kf = 0; kf < 32; ++kf) {
        const int kk = 4 * kf + 2 * h;
        v2f b0 = {0.f, 0.f}, b1 = {0.f, 0.f};
        if (c0.val) { b0.x = c0.wb[(size_t)kk * c0.ws]; b0.y = c0.wb[(size_t)(kk + 1) * c0.ws]; }
        if (c1.val) { b1.x = c1.wb[(size_t)kk * c1.ws]; b1.y = c1.wb[(size_t)(kk + 1) * c1.ws]; }
        B0[kf] = b0;
        B1[kf] = b1;
    }

    const int rt0 = blockIdx.x * TPB;

    // per-lane running output pointers (advance by 16 rows per tile)
    float* p0 = c0.ob + (size_t)(rt0 * 16 + 8 * h) * c0.os;
    float* p1 = c1.ob + (size_t)(rt0 * 16 + 8 * h) * c1.os;
    const int step0 = 16 * c0.os, step1 = 16 * c1.os;

    // ---- prologue: stage row tile 0 ----
    {
        const float* xt = x + (size_t)rt0 * 16 * KDIM;
#pragma unroll
        for (int j = 0; j < 4; ++j) {
            const int i = tid + j * 128, r = i >> 5, c4 = i & 31;
#if USE_ASYNC_LDS
            async_cp16(xt + r * KDIM + c4 * 4, &lds[0][r * PITCH + c4 * 4]);
#else
            *(v4f*)&lds[0][r * PITCH + c4 * 4] = *(const v4f*)(xt + r * KDIM + c4 * 4);
#endif
        }
    }
#if USE_ASYNC_LDS
    __builtin_amdgcn_s_wait_asynccnt(0);
#endif
    __syncthreads();

#pragma unroll 1
    for (int t = 0; t < TPB; ++t) {
        const int buf = t & 1;

        // kick off next tile's transfer early; overlaps with the WMMA work below
#if !USE_ASYNC_LDS
        v4f stage[4];
#endif
        if (t + 1 < TPB) {
            const float* xt = x + (size_t)(rt0 + t + 1) * 16 * KDIM;
#pragma unroll
            for (int j = 0; j < 4; ++j) {
                const int i = tid + j * 128, r = i >> 5, c4 = i & 31;
#if USE_ASYNC_LDS
                async_cp16(xt + r * KDIM + c4 * 4, &lds[buf ^ 1][r * PITCH + c4 * 4]);
#else
                stage[j] = *(const v4f*)(xt + r * KDIM + c4 * 4);
#endif
            }
        }

        // bias folded into accumulator init (all 8 C elements in a lane share column n)
        v8f acc0 = {c0.bias, c0.bias, c0.bias, c0.bias, c0.bias, c0.bias, c0.bias, c0.bias};
        v8f acc1 = {c1.bias, c1.bias, c1.bias, c1.bias, c1.bias, c1.bias, c1.bias, c1.bias};

        // K loop: 4-deep ds_load ring keeps LDS latency off the WMMA critical path.
        // A 16x4 f32: lane m = row; VGPR0/1 = K = 4kf+2h, 4kf+2h+1 -> one b64, all
        // 64 banks hit exactly once across the wave (pitch 132).
        const float* lbase = &lds[buf][m * PITCH + 2 * h];
        v2f abuf[4];
#pragma unroll
        for (int i = 0; i < 4; ++i) abuf[i] = *(const v2f*)(lbase + 4 * i);
#pragma unroll
        for (int kf = 0; kf < 32; ++kf) {
            const v2f a = abuf[kf & 3];
            if (kf < 28) abuf[kf & 3] = *(const v2f*)(lbase + 4 * (kf + 4));
            acc0 = wmma_f32x4(a, B0[kf], acc0);
            acc1 = wmma_f32x4(a, B1[kf], acc1);   // unconditional: EXEC stays all-1s
        }

        // store: D VGPR v holds (row = v + 8h, col = n); contiguous in n -> coalesced
#pragma unroll
        for (int v = 0; v < 8; ++v) {
            if (c0.val) p0[(size_t)v * c0.os] = acc0[v];
            if (c1.val) p1[(size_t)v * c1.os] = acc1[v];
        }
        p0 += step0;
        p1 += step1;

#if !USE_ASYNC_LDS
        if (t + 1 < TPB) {
#pragma unroll
            for (int j = 0; j < 4; ++j) {
                const int i = tid + j * 128, r = i >> 5, c4 = i & 31;
                *(v4f*)&lds[buf ^ 1][r * PITCH + c4 * 4] = stage[j];
            }
        }
#endif
#if USE_ASYNC_LDS
        __builtin_amdgcn_s_wait_asynccnt(0);
#endif
        __syncthreads();
    }
}

extern "C" void kernel_launch(void* const* d_in, const int* in_sizes, int n_in,
                              void* d_out, int out_size, void* d_ws, size_t ws_size,
                              hipStream_t stream) {
    // setup_inputs order: x, W_cls, b_cls, W_obj, b_obj, W_reg, b_reg (all fp32)
    const float* x  = (const float*)d_in[0];
    const float* Wc = (const float*)d_in[1];
    const float* bc = (const float*)d_in[2];
    const float* Wo = (const float*)d_in[3];
    const float* bo = (const float*)d_in[4];
    const float* Wr = (const float*)d_in[5];
    const float* br = (const float*)d_in[6];
    float* out = (float*)d_out;

    hipLaunchKernelGGL(anchor_head_fused, dim3(NBLK), dim3(128), 0, stream,
                       x, Wc, bc, Wo, bo, Wr, br, out);
}